// Model_39676907881634
// MI455X (gfx1250) — compile-verified
//
#include <hip/hip_runtime.h>

// Problem geometry
#define HW      307200      // 480*640
#define CHW3    921600      // 3*HW
#define CHW8    2457600     // 8*HW
#define NPIX    9830400     // 32*HW
#define NT4     2457600     // NPIX/4 (float4 tiles for stats)
#define NTILE2  307200      // NPIX/32 (pairs of 16-pixel WMMA tiles)
#define NBLK_STATS 2048
#define STATS_THREADS 256
#define EPS 1e-5

typedef float v2f __attribute__((ext_vector_type(2)));
typedef float v8f __attribute__((ext_vector_type(8)));

// ---------------------------------------------------------------------------
// Kernel 1: per-block partial sums of x moments:
//   acc[0..2] = sum x_c ;  acc[3..8] = sum x0x0, x0x1, x0x2, x1x1, x1x2, x2x2
// Deterministic: fixed grid-stride order per thread + fixed LDS tree reduce.
// Default (RT) loads so x1 is L2-resident for the transform pass.
// ---------------------------------------------------------------------------
__global__ __launch_bounds__(STATS_THREADS)
void stats_kernel(const float* __restrict__ x, float* __restrict__ partials) {
    float a0=0.f,a1=0.f,a2=0.f,a3=0.f,a4=0.f,a5=0.f,a6=0.f,a7=0.f,a8=0.f;
    int tid = (int)(blockIdx.x*blockDim.x + threadIdx.x);
    int nth = (int)(gridDim.x*blockDim.x);
    for (int t = tid; t < NT4; t += nth) {
        int p = t << 2;
        int b = p / HW;
        int s = p - b*HW;
        const float* xb = x + (size_t)b*CHW3 + s;
        float4 u = *(const float4*)(xb);
        float4 v = *(const float4*)(xb + HW);
        float4 w = *(const float4*)(xb + 2*HW);
        a0 += u.x+u.y+u.z+u.w;
        a1 += v.x+v.y+v.z+v.w;
        a2 += w.x+w.y+w.z+w.w;
        a3 += u.x*u.x + u.y*u.y + u.z*u.z + u.w*u.w;
        a4 += u.x*v.x + u.y*v.y + u.z*v.z + u.w*v.w;
        a5 += u.x*w.x + u.y*w.y + u.z*w.z + u.w*w.w;
        a6 += v.x*v.x + v.y*v.y + v.z*v.z + v.w*v.w;
        a7 += v.x*w.x + v.y*w.y + v.z*w.z + v.w*w.w;
        a8 += w.x*w.x + w.y*w.y + w.z*w.z + w.w*w.w;
    }
    __shared__ float red[STATS_THREADS];
    float acc[9] = {a0,a1,a2,a3,a4,a5,a6,a7,a8};
    for (int j = 0; j < 9; ++j) {
        __syncthreads();
        red[threadIdx.x] = acc[j];
        __syncthreads();
        for (int off = STATS_THREADS/2; off > 0; off >>= 1) {
            if ((int)threadIdx.x < off) red[threadIdx.x] += red[threadIdx.x + off];
            __syncthreads();
        }
        if (threadIdx.x == 0) partials[blockIdx.x*9 + j] = red[0];
    }
}

// ---------------------------------------------------------------------------
// Kernel 2: fold the whole network into out = M(8x3)*x + k(8).
// Single thread, FP64. BN(train) scale/shift derived from x moments:
//   mean(A x + c) = A*mx + c ;  var = diag(A Cov A^T)
// ---------------------------------------------------------------------------
__global__ void fold_kernel(const float* __restrict__ partials,
    const float* __restrict__ w1, const float* __restrict__ b1,
    const float* __restrict__ g2, const float* __restrict__ be2,
    const float* __restrict__ w3, const float* __restrict__ b3,
    const float* __restrict__ w4, const float* __restrict__ b4,
    const float* __restrict__ g5, const float* __restrict__ be5,
    const float* __restrict__ w6, const float* __restrict__ b6,
    const float* __restrict__ w7, const float* __restrict__ b7,
    const float* __restrict__ w8, const float* __restrict__ b8,
    float* __restrict__ mb)
{
    if (threadIdx.x != 0 || blockIdx.x != 0) return;
    double s[9];
    for (int j = 0; j < 9; ++j) s[j] = 0.0;
    for (int blk = 0; blk < NBLK_STATS; ++blk)
        for (int j = 0; j < 9; ++j) s[j] += (double)partials[blk*9 + j];

    const double N = (double)NPIX;
    double mx[3];
    for (int c = 0; c < 3; ++c) mx[c] = s[c] / N;
    double C[3][3];
    C[0][0] = s[3]/N - mx[0]*mx[0];
    C[0][1] = s[4]/N - mx[0]*mx[1];
    C[0][2] = s[5]/N - mx[0]*mx[2];
    C[1][1] = s[6]/N - mx[1]*mx[1];
    C[1][2] = s[7]/N - mx[1]*mx[2];
    C[2][2] = s[8]/N - mx[2]*mx[2];
    C[1][0] = C[0][1]; C[2][0] = C[0][2]; C[2][1] = C[1][2];

    // stage v1 = w1*x + b1 (3x3)
    double A[3][3], cc[3];
    for (int o = 0; o < 3; ++o) {
        for (int c = 0; c < 3; ++c) A[o][c] = (double)w1[o*3+c];
        cc[o] = (double)b1[o];
    }
    // BN1 (v2)
    for (int o = 0; o < 3; ++o) {
        double mu = cc[o];
        for (int c = 0; c < 3; ++c) mu += A[o][c]*mx[c];
        double var = 0.0;
        for (int c = 0; c < 3; ++c)
            for (int c2 = 0; c2 < 3; ++c2) var += A[o][c]*A[o][c2]*C[c][c2];
        double sc = (double)g2[o] / sqrt(var + EPS);
        double sh = (double)be2[o] - mu*sc;
        for (int c = 0; c < 3; ++c) A[o][c] *= sc;
        cc[o] = cc[o]*sc + sh;
    }
    // v3 = w3(8x3)*v2 + b3
    double A3[8][3], c3v[8];
    for (int p = 0; p < 8; ++p) {
        for (int c = 0; c < 3; ++c) {
            double acc = 0.0;
            for (int o = 0; o < 3; ++o) acc += (double)w3[p*3+o]*A[o][c];
            A3[p][c] = acc;
        }
        double acc = (double)b3[p];
        for (int o = 0; o < 3; ++o) acc += (double)w3[p*3+o]*cc[o];
        c3v[p] = acc;
    }
    // v4 = w4(8x8)*v3 + b4
    double A4[8][3], c4v[8];
    for (int q = 0; q < 8; ++q) {
        for (int c = 0; c < 3; ++c) {
            double acc = 0.0;
            for (int p = 0; p < 8; ++p) acc += (double)w4[q*8+p]*A3[p][c];
            A4[q][c] = acc;
        }
        double acc = (double)b4[q];
        for (int p = 0; p < 8; ++p) acc += (double)w4[q*8+p]*c3v[p];
        c4v[q] = acc;
    }
    // BN2 (v5)
    for (int q = 0; q < 8; ++q) {
        double mu = c4v[q];
        for (int c = 0; c < 3; ++c) mu += A4[q][c]*mx[c];
        double var = 0.0;
        for (int c = 0; c < 3; ++c)
            for (int c2 = 0; c2 < 3; ++c2) var += A4[q][c]*A4[q][c2]*C[c][c2];
        double sc = (double)g5[q] / sqrt(var + EPS);
        double sh = (double)be5[q] - mu*sc;
        for (int c = 0; c < 3; ++c) A4[q][c] *= sc;
        c4v[q] = c4v[q]*sc + sh;
    }
    // v6, v7, v8: three 8x8 affine stages
    const float* Ws[3] = {w6, w7, w8};
    const float* Bs[3] = {b6, b7, b8};
    double Ain[8][3], cin[8], Aout[8][3], cout[8];
    for (int q = 0; q < 8; ++q) { for (int c = 0; c < 3; ++c) Ain[q][c] = A4[q][c]; cin[q] = c4v[q]; }
    for (int stage = 0; stage < 3; ++stage) {
        const float* W = Ws[stage]; const float* B = Bs[stage];
        for (int q = 0; q < 8; ++q) {
            for (int c = 0; c < 3; ++c) {
                double acc = 0.0;
                for (int p = 0; p < 8; ++p) acc += (double)W[q*8+p]*Ain[p][c];
                Aout[q][c] = acc;
            }
            double acc = (double)B[q];
            for (int p = 0; p < 8; ++p) acc += (double)W[q*8+p]*cin[p];
            cout[q] = acc;
        }
        for (int q = 0; q < 8; ++q) { for (int c = 0; c < 3; ++c) Ain[q][c] = Aout[q][c]; cin[q] = cout[q]; }
    }
    for (int q = 0; q < 8; ++q) {
        for (int c = 0; c < 3; ++c) mb[q*3+c] = (float)Ain[q][c];
        mb[24+q] = (float)cin[q];
    }
}

// ---------------------------------------------------------------------------
// Kernel 3: out = M*x + k via V_WMMA_F32_16X16X4_F32, 2 tiles (32 px) / iter.
//   A (16x4): M = out channel (8 used), K = in channel (3 used)
//     lanes 0-15  : VGPR0=A[m,0], VGPR1=A[m,1]
//     lanes 16-31 : VGPR0=A[m,2], VGPR1=A[m,3]=0
//   B (4x16): N = pixel within 16-pixel tile
//     lanes 0-15  : VGPR0=x0[n], VGPR1=x1[n]
//     lanes 16-31 : VGPR0=x2[n], VGPR1=don't-care (A column K3 == 0)
//   C/D (16x16): VGPR r, lanes 0-15 -> out channel r, pixel n. Bias in C.
// Output stores are non-temporal so the 315 MB stream does not evict the
// 118 MB x1 working set (L2-resident from the stats pass) from the 192 MB L2.
// ---------------------------------------------------------------------------
__global__ __launch_bounds__(256)
void transform_kernel(const float* __restrict__ x,
                      const float* __restrict__ mb,
                      float* __restrict__ out)
{
    const int lane = (int)(threadIdx.x & 31);
    const int li   = lane & 15;
    const bool hi  = lane >= 16;

    v2f a; a.x = 0.0f; a.y = 0.0f;
    if (li < 8) {
        if (!hi) { a.x = mb[li*3+0]; a.y = mb[li*3+1]; }
        else     { a.x = mb[li*3+2]; }
    }
    v8f cacc;
#pragma unroll
    for (int r = 0; r < 8; ++r) cacc[r] = hi ? 0.0f : mb[24+r];

    const int selA = hi ? 2*HW : 0;     // K0 plane (low half) / K2 plane (high half)

    int wave  = (int)((blockIdx.x*blockDim.x + threadIdx.x) >> 5);
    int nwave = (int)((gridDim.x*blockDim.x) >> 5);
    for (int t = wave; t < NTILE2; t += nwave) {
        int p = t << 5;                 // 32 consecutive pixels; HW%32==0 so
        int b = p / HW;                 // the pair never straddles batch planes
        int s = p - b*HW;
        const float* xb = x + (size_t)b*CHW3 + s + li;
        // Two independent 16-pixel tiles -> 4 loads in flight before first wait
        v2f bm0, bm1;
        bm0.x = xb[selA];
        bm0.y = xb[HW];
        bm1.x = xb[selA + 16];
        bm1.y = xb[HW + 16];
        v8f d0 = __builtin_amdgcn_wmma_f32_16x16x4_f32(
                     false, a, false, bm0, (short)0, cacc, false, false);
        v8f d1 = __builtin_amdgcn_wmma_f32_16x16x4_f32(
                     false, a, false, bm1, (short)0, cacc, false, false);
        if (!hi) {
            float* ob = out + (size_t)b*CHW8 + s + li;
#pragma unroll
            for (int r = 0; r < 8; ++r) {
                __builtin_nontemporal_store(d0[r], ob + r*HW);
                __builtin_nontemporal_store(d1[r], ob + r*HW + 16);
            }
        }
    }
}

extern "C" void kernel_launch(void* const* d_in, const int* in_sizes, int n_in,
                              void* d_out, int out_size, void* d_ws, size_t ws_size,
                              hipStream_t stream) {
    const float* x1  = (const float*)d_in[0];
    const float* w1  = (const float*)d_in[1];
    const float* b1  = (const float*)d_in[2];
    const float* g2  = (const float*)d_in[3];
    const float* be2 = (const float*)d_in[4];
    const float* w3  = (const float*)d_in[5];
    const float* b3  = (const float*)d_in[6];
    const float* w4  = (const float*)d_in[7];
    const float* b4  = (const float*)d_in[8];
    const float* g5  = (const float*)d_in[9];
    const float* be5 = (const float*)d_in[10];
    const float* w6  = (const float*)d_in[11];
    const float* b6  = (const float*)d_in[12];
    const float* w7  = (const float*)d_in[13];
    const float* b7  = (const float*)d_in[14];
    const float* w8  = (const float*)d_in[15];
    const float* b8  = (const float*)d_in[16];

    float* ws       = (float*)d_ws;
    float* partials = ws;                      // NBLK_STATS*9 floats
    float* mb       = ws + NBLK_STATS*9;       // 24 (M) + 8 (bias) floats
    float* out      = (float*)d_out;

    stats_kernel<<<NBLK_STATS, STATS_THREADS, 0, stream>>>(x1, partials);
    fold_kernel<<<1, 64, 0, stream>>>(partials, w1, b1, g2, be2, w3, b3, w4, b4,
                                      g5, be5, w6, b6, w7, b7, w8, b8, mb);
    transform_kernel<<<2048, 256, 0, stream>>>(x1, mb, out);
}